// SpatialEncoder_29695403884713
// MI455X (gfx1250) — compile-verified
//
#include <hip/hip_runtime.h>

typedef __attribute__((ext_vector_type(2))) float v2f;
typedef __attribute__((ext_vector_type(8))) float v8f;

#define NN 100000
#define NE 1600000
#define D  128
#define EPSV 1e-5f

#define USE_ASYNC_LDS 1

// ---- gfx1250 async global->LDS copy (16B per lane), tracked by ASYNCcnt ----
__device__ __forceinline__ void g2l_b128(float* lds_dst, const float* gsrc) {
#if USE_ASYNC_LDS
  unsigned loff = (unsigned)(unsigned long long)lds_dst;  // low 32 bits used as the LDS byte address
  asm volatile("global_load_async_to_lds_b128 %0, %1, off"
               :: "v"(loff), "v"(gsrc) : "memory");
#else
  *(float4*)lds_dst = *(const float4*)gsrc;
#endif
}

__device__ __forceinline__ void wait_async_lds() {
#if USE_ASYNC_LDS
  asm volatile("s_wait_asynccnt 0x0" ::: "memory");
#endif
}

// ---------------- degree kernels ----------------
__global__ void gcn_deg_init(float* deg) {
  int i = blockIdx.x * blockDim.x + threadIdx.x;
  if (i < NN) deg[i] = 1.0f;  // self-loop
}

__global__ void gcn_deg_count(const int* __restrict__ dst, float* deg) {
  int e = blockIdx.x * blockDim.x + threadIdx.x;
  if (e < NE) atomicAdd(&deg[dst[e]], 1.0f);
}

__global__ void gcn_deg_rsqrt(float* deg) {
  int i = blockIdx.x * blockDim.x + threadIdx.x;
  if (i < NN) deg[i] = rsqrtf(deg[i]);
}

// ---- fp32 WMMA GEMM: Y = X @ W (+bias)(+relu), optional fused epilogue ----
// If aggOut != nullptr also writes aggOut = dinv[row]^2 * Y + bg2[col]
// (GCN self-loop term + conv bias), saving a full re-read pass over Y.
// One 256-thread block = 8 waves; each wave owns one 16x16 column tile.
__global__ __launch_bounds__(256) void gcn_gemm128_wmma(
    const float* __restrict__ X, const float* __restrict__ W,
    const float* __restrict__ bias, float* __restrict__ Y,
    const float* __restrict__ dinv, const float* __restrict__ bg2,
    float* __restrict__ aggOut, int numTiles, int do_relu)
{
  __shared__ float lw[D * D];    // 64 KB weights
  __shared__ float lx[16 * D];   // 8 KB A tile

  // stage W via async copies (completion enforced inside first tile iteration)
  for (int i = threadIdx.x; i < (D * D) / 4; i += 256)
    g2l_b128(&lw[i * 4], W + i * 4);

  const int wave = threadIdx.x >> 5;
  const int lane = threadIdx.x & 31;
  const int col0 = wave << 4;      // 8 waves x 16 cols = 128
  const int n    = lane & 15;      // col within tile (also A row m)
  const int hi   = lane >> 4;      // half-wave select
  const int kt   = hi << 1;        // K sub-offset: 0 or 2
  const float bv  = bias ? bias[col0 + n] : 0.0f;
  const float bgv = bg2 ? bg2[col0 + n] : 0.0f;

  for (int t = blockIdx.x; t < numTiles; t += gridDim.x) {
    const int row0 = t << 4;
    __syncthreads();  // all waves done reading previous A tile
    for (int i = threadIdx.x; i < (16 * D) / 4; i += 256)
      g2l_b128(&lx[i * 4], X + (size_t)row0 * D + i * 4);
    wait_async_lds();  // this wave's copies (incl. its share of W) landed
    __syncthreads();   // everyone's copies landed

    v8f c;
#pragma unroll
    for (int v = 0; v < 8; ++v) c[v] = bv;

#pragma unroll 8
    for (int k = 0; k < D; k += 4) {
      // A 16x4 fragment: lane m = lane&15 holds {K=kt, K=kt+1} in 2 VGPRs
      v2f a = *(const v2f*)&lx[n * D + k + kt];
      // B 4x16 fragment (mirrored layout): lane n holds rows {kt, kt+1}
      v2f b;
      b.x = lw[(k + kt) * D + col0 + n];
      b.y = lw[(k + kt + 1) * D + col0 + n];
      c = __builtin_amdgcn_wmma_f32_16x16x4_f32(
              /*neg_a=*/false, a, /*neg_b=*/false, b,
              /*c_mod=*/(short)0, c, /*reuse_a=*/false, /*reuse_b=*/false);
    }

    // C/D layout: VGPR v -> row (hi*8 + v), col = n
    float* yb = Y + (size_t)row0 * D + col0 + n;
    if (aggOut == nullptr) {
#pragma unroll
      for (int v = 0; v < 8; ++v) {
        float val = c[v];
        if (do_relu) val = fmaxf(val, 0.0f);
        yb[(size_t)(hi * 8 + v) * D] = val;
      }
    } else {
      float* ab = aggOut + (size_t)row0 * D + col0 + n;
#pragma unroll
      for (int v = 0; v < 8; ++v) {
        const int r = hi * 8 + v;
        float val = c[v];
        yb[(size_t)r * D] = val;
        float di = dinv[row0 + r];              // half-wave-uniform per v
        ab[(size_t)r * D] = di * di * val + bgv;
      }
    }
  }
}

// -------- edge scatter: agg[dst] += dinv[s]*dinv[d]*xw[src]; f4 per thread --
__global__ void gcn_edge_scatter(const int* __restrict__ src,
                                 const int* __restrict__ dst,
                                 const float* __restrict__ dinv,
                                 const float* __restrict__ xw,
                                 float* __restrict__ agg)
{
  unsigned idx = blockIdx.x * blockDim.x + threadIdx.x;  // < NE*32
  unsigned e = idx >> 5;
  unsigned q = (idx & 31) << 2;  // feature group of 4
  if (e < NE) {
    int s = src[e], d = dst[e];
    float norm = dinv[s] * dinv[d];
    float4 xv = *(const float4*)&xw[(size_t)s * D + q];
    float* ap = &agg[(size_t)d * D + q];
    atomicAdd(ap + 0, norm * xv.x);
    atomicAdd(ap + 1, norm * xv.y);
    atomicAdd(ap + 2, norm * xv.z);
    atomicAdd(ap + 3, norm * xv.w);
  }
}

// ---------------- out = relu(layernorm(agg + res)) ; one wave per node -----
__global__ __launch_bounds__(256) void gcn_post_ln(
    const float* __restrict__ agg, const float* __restrict__ res,
    const float* __restrict__ g, const float* __restrict__ b,
    float* __restrict__ out)
{
  int node = blockIdx.x * 8 + (threadIdx.x >> 5);
  int lane = threadIdx.x & 31;
  if (node >= NN) return;
  const float* ap = agg + (size_t)node * D;
  const float* rp = res + (size_t)node * D;
  float v[4];
  float s = 0.0f;
#pragma unroll
  for (int i = 0; i < 4; ++i) {
    v[i] = ap[lane + 32 * i] + rp[lane + 32 * i];
    s += v[i];
  }
#pragma unroll
  for (int o = 16; o; o >>= 1) s += __shfl_xor(s, o, 32);
  float mu = s * (1.0f / (float)D);
  float var = 0.0f;
#pragma unroll
  for (int i = 0; i < 4; ++i) { float d = v[i] - mu; var += d * d; }
#pragma unroll
  for (int o = 16; o; o >>= 1) var += __shfl_xor(var, o, 32);
  float rs = rsqrtf(var * (1.0f / (float)D) + EPSV);
  float* op = out + (size_t)node * D;
#pragma unroll
  for (int i = 0; i < 4; ++i) {
    int f = lane + 32 * i;
    op[f] = fmaxf((v[i] - mu) * rs * g[f] + b[f], 0.0f);
  }
}

extern "C" void kernel_launch(void* const* d_in, const int* in_sizes, int n_in,
                              void* d_out, int out_size, void* d_ws, size_t ws_size,
                              hipStream_t stream) {
  (void)in_sizes; (void)n_in; (void)out_size; (void)ws_size;

  const float* nf = (const float*)d_in[0];   // [N, 128]
  const int*   ei = (const int*)d_in[1];     // [2, E]
  const float* Wp = (const float*)d_in[2];   // [128, 128]
  const float* bp = (const float*)d_in[3];   // [128]
  const float* Wg = (const float*)d_in[4];   // [2, 128, 128]
  const float* bg = (const float*)d_in[5];   // [2, 128]
  const float* lg = (const float*)d_in[6];   // [128]
  const float* lb = (const float*)d_in[7];   // [128]
  float* out = (float*)d_out;                // [N, 128]

  const int* srcp = ei;
  const int* dstp = ei + NE;

  char* ws = (char*)d_ws;
  float* dinv = (float*)ws;                        // N floats
  float* x    = (float*)(ws + ((size_t)NN * 4));   // N*D (16B aligned)
  float* xw   = x + (size_t)NN * D;
  float* agg  = xw + (size_t)NN * D;

  const int numTiles = NN / 16;  // 6250 exact

  // degrees -> dinv_sqrt (in place)
  gcn_deg_init<<<(NN + 255) / 256, 256, 0, stream>>>(dinv);
  gcn_deg_count<<<(NE + 255) / 256, 256, 0, stream>>>(dstp, dinv);
  gcn_deg_rsqrt<<<(NN + 255) / 256, 256, 0, stream>>>(dinv);

  // x = relu(nf @ W_proj + b_proj)
  gcn_gemm128_wmma<<<2048, 256, 0, stream>>>(nf, Wp, bp, x,
                                             nullptr, nullptr, nullptr,
                                             numTiles, 1);

  for (int l = 0; l < 2; ++l) {
    // xw = x @ W_gcn[l] ; fused: agg = dinv^2 * xw + b_gcn[l]
    gcn_gemm128_wmma<<<2048, 256, 0, stream>>>(x, Wg + (size_t)l * D * D,
                                               nullptr, xw,
                                               dinv, bg + l * D, agg,
                                               numTiles, 0);
    // agg[dst] += dinv[src]*dinv[dst] * xw[src]
    gcn_edge_scatter<<<(NE * 32 + 255) / 256, 256, 0, stream>>>(
        srcp, dstp, dinv, xw, agg);
    // x/out = relu(layernorm(agg + x))
    gcn_post_ln<<<(NN + 7) / 8, 256, 0, stream>>>(agg, x, lg, lb,
                                                  (l == 1) ? out : x);
  }
}